// Encoder_71030169141526
// MI455X (gfx1250) — compile-verified
//
#include <hip/hip_runtime.h>
#include <hip/hip_bf16.h>

// ---------------------------------------------------------------------------
// 2-layer GCN: norm -> (x@W1, gather/scale/scatter, +b1, relu) -> (h@W2, ... , +b2)
// GEMMs use V_WMMA_F32_16X16X4_F32 (fp32 exact). Scatter uses per-wave
// coalesced global_atomic_add_f32.
// ---------------------------------------------------------------------------

typedef float v2f __attribute__((ext_vector_type(2)));
typedef float v8f __attribute__((ext_vector_type(8)));

#define N_NODES 100000
#define IN_CH   128
#define HID     64
#define OUT_CH  32

// ---- init: deg = 1 (self loop), agg1 = 0, out = 0 --------------------------
__global__ void k_init(float* __restrict__ deg, float* __restrict__ agg1,
                       float* __restrict__ out, int n) {
  long i = (long)blockIdx.x * blockDim.x + threadIdx.x;
  if (i < n) deg[i] = 1.0f;                 // self-loop contributes 1 to degree
  if (i < (long)n * HID) agg1[i] = 0.0f;
  if (i < (long)n * OUT_CH) out[i] = 0.0f;
}

// ---- degree accumulation over real edges -----------------------------------
__global__ void k_deg(const int* __restrict__ dst, float* __restrict__ deg, int e) {
  int i = blockIdx.x * blockDim.x + threadIdx.x;
  if (i < e) {
    (void)__hip_atomic_fetch_add(&deg[dst[i]], 1.0f,
                                 __ATOMIC_RELAXED, __HIP_MEMORY_SCOPE_AGENT);
  }
}

// ---- dinv = rsqrt(deg) in place (deg >= 1 always due to self loops) --------
__global__ void k_rsqrt(float* __restrict__ d, int n) {
  int i = blockIdx.x * blockDim.x + threadIdx.x;
  if (i < n) d[i] = rsqrtf(d[i]);
}

// ---- fp32 WMMA GEMM: out[M,NC] = A[M,K] @ W[K,NC]; one wave per 16x16 tile -
__global__ void k_gemm_f32(const float* __restrict__ A, const float* __restrict__ W,
                           float* __restrict__ out, int M, int K, int NC,
                           int tiles_n, int total_tiles) {
  int wave = (int)((blockIdx.x * (long)blockDim.x + threadIdx.x) >> 5);
  if (wave >= total_tiles) return;          // wave-uniform: EXEC stays all-1s
  int lane  = threadIdx.x & 31;
  int tm    = wave / tiles_n;
  int tn    = wave % tiles_n;
  int hi    = lane >> 4;                    // 0: K pair {0,1}, 1: K pair {2,3}
  int l15   = lane & 15;
  int row   = tm * 16 + l15;                // A row for this lane
  int col   = tn * 16 + l15;                // B/C column for this lane
  int rrow  = row < M ? row : M - 1;        // clamp (N is a multiple of 16 here)

  const float* arow = A + (size_t)rrow * K + 2 * hi;
  const float* wcol = W + (size_t)(2 * hi) * NC + col;

  v8f c = {};
  for (int k0 = 0; k0 < K; k0 += 4) {
    v2f a = *(const v2f*)(arow + k0);                    // global_load_b64
    v2f b;
    b.x = wcol[(size_t)k0 * NC];
    b.y = wcol[(size_t)(k0 + 1) * NC];
    c = __builtin_amdgcn_wmma_f32_16x16x4_f32(
        /*neg_a=*/false, a, /*neg_b=*/false, b,
        /*c_mod=*/(short)0, c, /*reuse_a=*/false, /*reuse_b=*/false);
  }

  // C/D layout: VGPR v -> M = v (lanes 0-15) or M = v + 8 (lanes 16-31)
  int m0 = tm * 16 + hi * 8;
  #pragma unroll
  for (int v = 0; v < 8; ++v) {
    int r = m0 + v;
    if (r < M) out[(size_t)r * NC + col] = c[v];
  }
}

// ---- edge scatter: agg[dst] += m[src] * dinv[src]*dinv[dst]; wave per edge -
template <int C>
__global__ void k_scatter(const float* __restrict__ m, const int* __restrict__ src,
                          const int* __restrict__ dst, const float* __restrict__ dinv,
                          float* __restrict__ agg, int e) {
  int wave = (int)((blockIdx.x * (long)blockDim.x + threadIdx.x) >> 5);
  if (wave >= e) return;
  int lane = threadIdx.x & 31;
  int s = src[wave];
  int d = dst[wave];
  float w = dinv[s] * dinv[d];              // wave-uniform scalar
  #pragma unroll
  for (int c = lane; c < C; c += 32) {
    float val = m[(size_t)s * C + c] * w;   // coalesced gather
    (void)__hip_atomic_fetch_add(&agg[(size_t)d * C + c], val,
                                 __ATOMIC_RELAXED, __HIP_MEMORY_SCOPE_AGENT);
  }
}

// ---- layer-1 epilogue: h = relu(agg + m*dinv^2 + b1) (in place on agg) -----
__global__ void k_post1(float* __restrict__ agg, const float* __restrict__ m,
                        const float* __restrict__ dinv, const float* __restrict__ b1,
                        int n) {
  long i = (long)blockIdx.x * blockDim.x + threadIdx.x;
  if (i >= (long)n * HID) return;
  int node = (int)(i >> 6);                 // HID == 64
  int ch   = (int)(i & 63);
  float di = dinv[node];
  float v  = agg[i] + m[i] * di * di + b1[ch];
  agg[i]   = v > 0.0f ? v : 0.0f;
}

// ---- layer-2 epilogue: out += m2*dinv^2 + b2 (no relu) ---------------------
__global__ void k_post2(float* __restrict__ out, const float* __restrict__ m2,
                        const float* __restrict__ dinv, const float* __restrict__ b2,
                        int n) {
  long i = (long)blockIdx.x * blockDim.x + threadIdx.x;
  if (i >= (long)n * OUT_CH) return;
  int node = (int)(i >> 5);                 // OUT_CH == 32
  int ch   = (int)(i & 31);
  float di = dinv[node];
  out[i] = out[i] + m2[i] * di * di + b2[ch];
}

// ---------------------------------------------------------------------------
extern "C" void kernel_launch(void* const* d_in, const int* in_sizes, int n_in,
                              void* d_out, int out_size, void* d_ws, size_t ws_size,
                              hipStream_t stream) {
  const float* x   = (const float*)d_in[0];     // [N, 128]
  const int*   ei  = (const int*)d_in[1];       // [2, E] (int32 per harness)
  const float* W1  = (const float*)d_in[2];     // [128, 64]
  const float* b1  = (const float*)d_in[3];     // [64]
  const float* W2  = (const float*)d_in[4];     // [64, 32]
  const float* b2  = (const float*)d_in[5];     // [32]
  float*       out = (float*)d_out;             // [N, 32]

  const int N = in_sizes[0] / IN_CH;
  const int E = in_sizes[1] / 2;
  const int* src = ei;
  const int* dst = ei + E;

  // workspace layout (floats): dinv | bufA (m1 / m2) | bufB (agg1 / h)
  float* dinv = (float*)d_ws;
  float* bufA = dinv + 131072;                  // pad to 512 KB
  float* bufB = bufA + (size_t)N * HID;

  const int BT = 256;

  // 1) init deg=1, agg1=0, out=0
  {
    long total = (long)N * HID;
    k_init<<<(int)((total + BT - 1) / BT), BT, 0, stream>>>(dinv, bufB, out, N);
  }
  // 2) degree accumulation
  k_deg<<<(E + BT - 1) / BT, BT, 0, stream>>>(dst, dinv, E);
  // 3) dinv = rsqrt(deg)
  k_rsqrt<<<(N + BT - 1) / BT, BT, 0, stream>>>(dinv, N);
  // 4) m1 = x @ W1   (WMMA fp32)
  {
    int tiles_m = (N + 15) / 16, tiles_n = HID / 16;
    int tiles = tiles_m * tiles_n;
    long threads = (long)tiles * 32;
    k_gemm_f32<<<(int)((threads + BT - 1) / BT), BT, 0, stream>>>(
        x, W1, bufA, N, IN_CH, HID, tiles_n, tiles);
  }
  // 5) scatter layer 1: agg1 += m1[src]*norm
  {
    long threads = (long)E * 32;
    k_scatter<HID><<<(int)((threads + BT - 1) / BT), BT, 0, stream>>>(
        bufA, src, dst, dinv, bufB, E);
  }
  // 6) h = relu(agg1 + m1*dinv^2 + b1)
  {
    long total = (long)N * HID;
    k_post1<<<(int)((total + BT - 1) / BT), BT, 0, stream>>>(bufB, bufA, dinv, b1, N);
  }
  // 7) m2 = h @ W2   (WMMA fp32), reuse bufA
  {
    int tiles_m = (N + 15) / 16, tiles_n = OUT_CH / 16;
    int tiles = tiles_m * tiles_n;
    long threads = (long)tiles * 32;
    k_gemm_f32<<<(int)((threads + BT - 1) / BT), BT, 0, stream>>>(
        bufB, W2, bufA, N, HID, OUT_CH, tiles_n, tiles);
  }
  // 8) scatter layer 2 into out
  {
    long threads = (long)E * 32;
    k_scatter<OUT_CH><<<(int)((threads + BT - 1) / BT), BT, 0, stream>>>(
        bufA, src, dst, dinv, out, E);
  }
  // 9) out += m2*dinv^2 + b2
  {
    long total = (long)N * OUT_CH;
    k_post2<<<(int)((total + BT - 1) / BT), BT, 0, stream>>>(out, bufA, dinv, b2, N);
  }
}